// GNN_66580583022905
// MI455X (gfx1250) — compile-verified
//
#include <hip/hip_runtime.h>
#include <hip/hip_bf16.h>

typedef __attribute__((ext_vector_type(2))) float v2f;
typedef __attribute__((ext_vector_type(8))) float v8f;

#define TPB 256

// ---------------------------------------------------------------- utilities
__global__ __launch_bounds__(TPB) void fill_f32(float* __restrict__ p, float v, long n) {
    long i = (long)blockIdx.x * TPB + threadIdx.x;
    if (i < n) p[i] = v;
}

__global__ __launch_bounds__(TPB) void degree_accum(const int* __restrict__ col,
                                                    float* __restrict__ deg, int nedges) {
    int e = blockIdx.x * TPB + threadIdx.x;
    if (e < nedges) atomicAdd(&deg[col[e]], 1.0f);
}

__global__ __launch_bounds__(TPB) void rsqrt_inplace(float* __restrict__ d, int n) {
    int i = blockIdx.x * TPB + threadIdx.x;
    if (i < n) d[i] = rsqrtf(d[i]);   // deg >= 1 always (self-loops)
}

// ---------------------------------------------------------------- WMMA GEMM
// h[nrows x 64] = x[nrows x 64] @ W[64 x 64], full f32 via V_WMMA_F32_16X16X4_F32.
// One wave (32 lanes) computes one 16x16 tile of h; 8 waves / block.
__global__ __launch_bounds__(TPB) void gemm64_wmma_f32(const float* __restrict__ x,
                                                       const float* __restrict__ W,
                                                       float* __restrict__ h, int nrows) {
    const int wave = threadIdx.x >> 5;
    const int lane = threadIdx.x & 31;
    const int half = lane >> 4;     // 0: lanes 0-15, 1: lanes 16-31
    const int lrow = lane & 15;

    const int mtiles = (nrows + 15) >> 4;
    const int tile   = blockIdx.x * 8 + wave;
    const int mt     = tile >> 2;   // row-tile
    const int nt     = tile & 3;    // col-tile (64/16 = 4)
    if (mt >= mtiles) return;

    const int m0 = mt << 4;
    const int n0 = nt << 4;

    // clamp row index for safe loads on a ragged tail (N=100000 is 16-aligned anyway)
    int mr = m0 + lrow;
    if (mr >= nrows) mr = nrows - 1;

    const float* __restrict__ xrow = x + (size_t)mr * 64;
    const float* __restrict__ wcol = W + n0 + lrow;

    v8f c = {};
#pragma unroll
    for (int k0 = 0; k0 < 64; k0 += 4) {
        // A 16x4 f32 layout: VGPR0 = K(2*half), VGPR1 = K(2*half+1), row = lrow
        v2f a;
        a.x = xrow[k0 + 2 * half + 0];
        a.y = xrow[k0 + 2 * half + 1];
        // B 4x16 f32 layout (mirrored): row K striped across lanes within a VGPR
        v2f b;
        b.x = wcol[(size_t)(k0 + 2 * half + 0) * 64];
        b.y = wcol[(size_t)(k0 + 2 * half + 1) * 64];
        c = __builtin_amdgcn_wmma_f32_16x16x4_f32(false, a, false, b,
                                                  (short)0, c, false, false);
    }

    // C/D 16x16 f32: VGPR r -> (M = r + 8*half, N = lrow)
    float* __restrict__ hbase = h + (size_t)m0 * 64 + n0;
#pragma unroll
    for (int r = 0; r < 8; ++r) {
        int m = r + 8 * half;
        if (m0 + m < nrows) hbase[(size_t)m * 64 + lrow] = c[r];
    }
}

// ---------------------------------------------------------------- edge scatter
// 16 lanes per edge, float4 per lane (GLOBAL_LOAD_B128 gather + 4 f32 atomics).
__global__ __launch_bounds__(TPB) void edge_scatter(const int* __restrict__ row,
                                                    const int* __restrict__ col,
                                                    const float* __restrict__ dinv,
                                                    const float* __restrict__ h,
                                                    float* __restrict__ acc, int nedges) {
    int t = blockIdx.x * TPB + threadIdx.x;
    int e = t >> 4;
    int k = (t & 15) << 2;               // column chunk k..k+3
    if (e >= nedges) return;

    const int r   = row[e];
    const int c   = col[e];
    const float s = dinv[r] * dinv[c];

    const float4 v = *(const float4*)(h + (size_t)r * 64 + k);
    float* __restrict__ dst = acc + (size_t)c * 64 + k;
    atomicAdd(dst + 0, v.x * s);
    atomicAdd(dst + 1, v.y * s);
    atomicAdd(dst + 2, v.z * s);
    atomicAdd(dst + 3, v.w * s);
}

// ---------------------------------------------------------------- epilogue
// out = acc + h[i]*dinv[i]^2 (self-loop, no atomics needed) + bias, optional relu.
__global__ __launch_bounds__(TPB) void epilogue(const float* __restrict__ acc,
                                                const float* __restrict__ h,
                                                const float* __restrict__ dinv,
                                                const float* __restrict__ bias,
                                                float* __restrict__ out, int n, int do_relu) {
    int t = blockIdx.x * TPB + threadIdx.x;
    int i = t >> 4;
    int k = (t & 15) << 2;
    if (i >= n) return;

    const float di = dinv[i];
    const float s  = di * di;
    const float4 a  = *(const float4*)(acc  + (size_t)i * 64 + k);
    const float4 hv = *(const float4*)(h    + (size_t)i * 64 + k);
    const float4 b4 = *(const float4*)(bias + k);

    float4 o;
    o.x = a.x + hv.x * s + b4.x;
    o.y = a.y + hv.y * s + b4.y;
    o.z = a.z + hv.z * s + b4.z;
    o.w = a.w + hv.w * s + b4.w;
    if (do_relu) {
        o.x = fmaxf(o.x, 0.f); o.y = fmaxf(o.y, 0.f);
        o.z = fmaxf(o.z, 0.f); o.w = fmaxf(o.w, 0.f);
    }
    *(float4*)(out + (size_t)i * 64 + k) = o;
}

// ---------------------------------------------------------------- launcher
static inline int cdiv(long a, long b) { return (int)((a + b - 1) / b); }

extern "C" void kernel_launch(void* const* d_in, const int* in_sizes, int n_in,
                              void* d_out, int out_size, void* d_ws, size_t ws_size,
                              hipStream_t stream) {
    const int*   edge_index = (const int*)d_in[0];
    const int    E  = in_sizes[0] / 2;
    const float* emb = (const float*)d_in[1];
    const int    N  = in_sizes[1] / 64;
    const float* W1 = (const float*)d_in[2];
    const float* b1 = (const float*)d_in[3];
    const float* W2 = (const float*)d_in[4];
    const float* b2 = (const float*)d_in[5];
    float*       out = (float*)d_out;

    const int* rowp = edge_index;
    const int* colp = edge_index + E;

    // workspace layout: dinv[N] | h[N*64] | acc[N*64]  (~52 MB @ N=100k)
    float* ws = (float*)d_ws;
    const size_t nd   = (size_t)N * 64;
    const size_t offh = ((size_t)N + 63) & ~(size_t)63;
    float* dinv = ws;
    float* h    = ws + offh;
    float* acc  = h + nd;

    const int mtiles = (N + 15) >> 4;
    const int gemm_blocks = cdiv((long)mtiles * 4, 8);

    // --- normalization ---
    fill_f32<<<cdiv(N, TPB), TPB, 0, stream>>>(dinv, 1.0f, N);      // self-loop degree
    fill_f32<<<cdiv(nd, TPB), TPB, 0, stream>>>(acc, 0.0f, (long)nd);
    degree_accum<<<cdiv(E, TPB), TPB, 0, stream>>>(colp, dinv, E);
    rsqrt_inplace<<<cdiv(N, TPB), TPB, 0, stream>>>(dinv, N);

    // --- layer 1 ---
    gemm64_wmma_f32<<<gemm_blocks, TPB, 0, stream>>>(emb, W1, h, N);
    edge_scatter<<<cdiv((long)E * 16, TPB), TPB, 0, stream>>>(rowp, colp, dinv, h, acc, E);
    epilogue<<<cdiv((long)N * 16, TPB), TPB, 0, stream>>>(acc, h, dinv, b1, acc, N, 1); // in-place -> x2

    // --- layer 2 ---
    gemm64_wmma_f32<<<gemm_blocks, TPB, 0, stream>>>(acc, W2, h, N);
    fill_f32<<<cdiv(nd, TPB), TPB, 0, stream>>>(acc, 0.0f, (long)nd);
    edge_scatter<<<cdiv((long)E * 16, TPB), TPB, 0, stream>>>(rowp, colp, dinv, h, acc, E);
    epilogue<<<cdiv((long)N * 16, TPB), TPB, 0, stream>>>(acc, h, dinv, b2, out, N, 0);
}